// CoInHead_31628139168020
// MI455X (gfx1250) — compile-verified
//
#include <hip/hip_runtime.h>
#include <hip/hip_bf16.h>

// ---------------- problem constants (match reference) ----------------
#define BATCH        16
#define NUM_MAX_OBJS 500
#define NUM_CLASSES  10
#define FW           512
#define FH           512
#define RMAX         16
#define MIN_RADIUS   2
#define OVERLAP      0.1f
#define PCR0f        (-51.2f)
#define PCR1f        (-51.2f)
#define VXSf         (0.05f * 4.0f)    // VX * STRIDE
#define VYSf         (0.05f * 4.0f)
#define EPSF         1.1920929e-07f    // float32 eps

#define HEAT_ELEMS   ((size_t)BATCH * NUM_CLASSES * FH * FW)   // 41,943,040
#define RB_ELEMS     ((size_t)BATCH * NUM_MAX_OBJS * 8)        // 64,000
#define NOBJ         (BATCH * NUM_MAX_OBJS)                    // 8,000

typedef __attribute__((ext_vector_type(2))) float v2f;
typedef __attribute__((ext_vector_type(8))) float v8f;

__device__ __forceinline__ float gaussian_radius(float h, float w, float mo) {
    float b1 = h + w;
    float c1 = w * h * (1.f - mo) / (1.f + mo);
    float sq1 = sqrtf(fmaxf(b1 * b1 - 4.f * c1, 0.f));
    float r1 = (b1 + sq1) * 0.5f;
    float b2 = 2.f * (h + w);
    float c2 = (1.f - mo) * w * h;
    float sq2 = sqrtf(fmaxf(b2 * b2 - 16.f * c2, 0.f));
    float r2 = (b2 + sq2) * 0.5f;
    float a3 = 4.f * mo;
    float b3 = -2.f * mo * (h + w);
    float c3 = (mo - 1.f) * w * h;
    float sq3 = sqrtf(fmaxf(b3 * b3 - 4.f * a3 * c3, 0.f));
    float r3 = (b3 + sq3) / (2.f * a3);
    return fminf(fminf(r1, r2), r3);
}

// ---------------- kernel 1: zero the heatmap (168 MB, HBM-bound floor) ----------------
__global__ __launch_bounds__(256) void zero_heat_kernel(float4* __restrict__ p) {
    size_t i = (size_t)blockIdx.x * 256 + threadIdx.x;   // HEAT_ELEMS/4 = 10,485,760 exact
    p[i] = make_float4(0.f, 0.f, 0.f, 0.f);
}

// ---------------- kernel 2: per-object Gaussian splat via rank-1 WMMA ----------------
// One wave32 per object. Separable gaussian: g = a (x) b via V_WMMA_F32_16X16X4_F32
// (a in the K=0 column of A, b in the K=0 row of B, all other K slots zero).
// All per-object scalars are forced into SGPRs via readfirstlane so the tile
// skip logic and the atomic base address are scalar (s_cbranch, saddr atomics).
__global__ __launch_bounds__(32) void splat_kernel(const float* __restrict__ gt,
                                                   float* __restrict__ heat) {
    const int obj  = blockIdx.x;            // 0 .. NOBJ-1
    const int b    = obj / NUM_MAX_OBJS;
    const int lane = threadIdx.x;           // 0 .. 31 (wave32)

    // broadcast load of the 8 box scalars (same address across all lanes)
    const float* box = gt + (size_t)obj * 8;
    const float x    = box[0];
    const float y    = box[1];
    const float dx   = box[3];
    const float dy   = box[4];
    const float clsf = box[7];

    const float coord_x = fminf(fmaxf((x - PCR0f) / VXSf, 0.f), (float)FW - 0.5f);
    const float coord_y = fminf(fmaxf((y - PCR1f) / VYSf, 0.f), (float)FH - 0.5f);
    const float dxf = dx / VXSf;
    const float dyf = dy / VYSf;

    // wave-uniform validity -> scalar branch (no exec masking before the WMMAs)
    const int valid = (dxf > 0.f) && (dyf > 0.f);
    if (__builtin_amdgcn_readfirstlane(valid) == 0) return;

    int r = (int)gaussian_radius(dxf, dyf, OVERLAP);
    r = min(max(r, MIN_RADIUS), RMAX);
    int cls = (int)(clsf - 1.f);
    cls = min(max(cls, 0), NUM_CLASSES - 1);

    // scalarize all wave-uniform quantities
    const int r_s   = __builtin_amdgcn_readfirstlane(r);
    const int cxi_s = __builtin_amdgcn_readfirstlane((int)coord_x);
    const int cyi_s = __builtin_amdgcn_readfirstlane((int)coord_y);
    const int cls_s = __builtin_amdgcn_readfirstlane(cls);

    const float sigma  = (2.f * (float)r_s + 1.f) / 6.f;
    const float inv2s2 = 1.f / (2.f * sigma * sigma);

    // 1D gaussian, one sample per lane: e[t] = exp(-(t-16)^2 / (2 sigma^2)).
    // e[32] == e[0] by symmetry, so lane (m & 31) covers m = 0..32.
    const float offt = (float)(lane - RMAX);
    const float e1d  = __expf(-offt * offt * inv2s2);

    float* plane = heat + ((size_t)b * NUM_CLASSES + cls_s) * (size_t)(FH * FW);

    // Rows/cols needed: [16-r, 16+r]. Tile t (span 16t..16t+15) is needed iff
    // 16t <= 16+r, i.e. tile index 2 only when r == 16. thi = 1 or 2 (scalar).
    const int thi = (RMAX + r_s) >> 4;

    // 33x33 patch padded to 48x48 => up to 3x3 tiles of 16x16.
    #pragma unroll
    for (int ti = 0; ti < 3; ++ti) {
        if (ti > thi) break;                       // scalar skip of unneeded row tiles
        // A fragment (16x4 f32, 2 VGPRs/lane): a[m] at K=0 (lanes 0-15, vgpr0); else 0.
        const int mrow = ti * 16 + lane;
        float aval = __shfl(e1d, mrow & 31, 32);
        if (lane >= 16 || mrow > 32) aval = 0.f;
        v2f afrag; afrag.x = aval; afrag.y = 0.f;

        #pragma unroll
        for (int tj = 0; tj < 3; ++tj) {
            if (tj > thi) break;                   // scalar skip of unneeded col tiles
            // B fragment (4x16 f32): b[n] at K=0 (lanes 0-15, vgpr0); else 0.
            const int ncol = tj * 16 + lane;
            float bval = __shfl(e1d, ncol & 31, 32);
            if (lane >= 16 || ncol > 32) bval = 0.f;
            v2f bfrag; bfrag.x = bval; bfrag.y = 0.f;

            v8f acc = {};
            // D = A x B + 0 : rank-1 outer product tile g[m][n] = a[m]*b[n]
            acc = __builtin_amdgcn_wmma_f32_16x16x4_f32(
                /*neg_a=*/false, afrag, /*neg_b=*/false, bfrag,
                /*c_mod=*/(short)0, acc, /*reuse_a=*/false, /*reuse_b=*/false);

            // C/D layout: VGPR v, lanes 0-15 -> M=v, N=lane; lanes 16-31 -> M=v+8, N=lane-16
            const int n  = (lane & 15) + tj * 16;
            const int oj = n - RMAX;
            const int xx = cxi_s + oj;
            const bool colok = (n <= 32) && (oj >= -r_s) && (oj <= r_s) && (xx >= 0) && (xx < FW);
            if (colok) {
                const int mbase = ti * 16 + ((lane >> 4) << 3);
                #pragma unroll
                for (int v = 0; v < 8; ++v) {
                    const int m  = mbase + v;
                    const int oi = m - RMAX;
                    const int yy = cyi_s + oi;
                    const float g = acc[v];
                    if (m <= 32 && oi >= -r_s && oi <= r_s && yy >= 0 && yy < FH && g >= EPSF) {
                        // float-max via int-max: all values >= 0, IEEE order == int order.
                        atomicMax((int*)(plane + yy * FW + xx), __float_as_int(g));
                    }
                }
            }
        }
    }
}

// ---------------- kernel 3: ret_boxes / inds / mask ----------------
__global__ __launch_bounds__(256) void boxes_kernel(const float* __restrict__ gt,
                                                    float* __restrict__ rb,
                                                    long long* __restrict__ inds,
                                                    long long* __restrict__ mask) {
    const int i = blockIdx.x * 256 + threadIdx.x;
    if (i >= NOBJ) return;
    const float* box = gt + (size_t)i * 8;
    const float x = box[0], y = box[1], z = box[2];
    const float dx = box[3], dy = box[4], dz = box[5];
    const float heading = box[6];

    const float coord_x = fminf(fmaxf((x - PCR0f) / VXSf, 0.f), (float)FW - 0.5f);
    const float coord_y = fminf(fmaxf((y - PCR1f) / VYSf, 0.f), (float)FH - 0.5f);
    const int   cxi = (int)coord_x;
    const int   cyi = (int)coord_y;
    const float dxf = dx / VXSf;
    const float dyf = dy / VYSf;
    const bool valid = (dxf > 0.f) && (dyf > 0.f);
    const float vf = valid ? 1.f : 0.f;

    float* o = rb + (size_t)i * 8;
    o[0] = (coord_x - (float)cxi) * vf;
    o[1] = (coord_y - (float)cyi) * vf;
    o[2] = z * vf;
    o[3] = logf(fmaxf(dx, 1e-12f)) * vf;
    o[4] = logf(fmaxf(dy, 1e-12f)) * vf;
    o[5] = logf(fmaxf(dz, 1e-12f)) * vf;
    o[6] = cosf(heading) * vf;
    o[7] = sinf(heading) * vf;

    inds[i] = valid ? (long long)(cyi * FW + cxi) : 0ll;
    mask[i] = valid ? 1ll : 0ll;
}

extern "C" void kernel_launch(void* const* d_in, const int* in_sizes, int n_in,
                              void* d_out, int out_size, void* d_ws, size_t ws_size,
                              hipStream_t stream) {
    (void)in_sizes; (void)n_in; (void)d_ws; (void)ws_size; (void)out_size;

    const float* gt = (const float*)d_in[0];

    // d_out layout: heatmap f32 | ret_boxes f32 | inds i64 | mask i64
    float*     heat = (float*)d_out;
    float*     rb   = heat + HEAT_ELEMS;
    long long* inds = (long long*)(rb + RB_ELEMS);
    long long* mask = inds + NOBJ;

    // 1) zero the heatmap (10,485,760 float4 stores)
    zero_heat_kernel<<<(unsigned)(HEAT_ELEMS / 4 / 256), 256, 0, stream>>>((float4*)heat);

    // 2) per-object regression targets
    boxes_kernel<<<(NOBJ + 255) / 256, 256, 0, stream>>>(gt, rb, inds, mask);

    // 3) gaussian splat: one wave32 per object, WMMA rank-1 outer product
    splat_kernel<<<NOBJ, 32, 0, stream>>>(gt, heat);
}